// AlignModule_25494925869466
// MI455X (gfx1250) — compile-verified
//
#include <hip/hip_runtime.h>
#include <hip/hip_bf16.h>
#include <math.h>

// Problem constants (from reference): B=32, C=512, H=24, W=8 -> hw=192, C8=64
#define BB    32
#define CC    512
#define HW    192
#define C8    64

typedef __attribute__((ext_vector_type(2))) float v2f;
typedef __attribute__((ext_vector_type(4))) float v4f;
typedef __attribute__((ext_vector_type(8))) float v8f;

#define KCH      32          // K-chunk staged in LDS
#define LSTRIDE  36          // K-minor stride: pos*36 mod 64 -> 16 residues ==0 mod 4;
                             // upper half (+2) -> ==2 mod 4: b64 reads are bank-conflict-free

static __device__ __forceinline__ v8f wmma_f32_k4(v2f a, v2f b, v8f c) {
    return __builtin_amdgcn_wmma_f32_16x16x4_f32(
        /*neg_a=*/false, a, /*neg_b=*/false, b,
        /*c_mod=*/(short)0, c, /*reuse_a=*/false, /*reuse_b=*/false);
}

// ---------------------------------------------------------------------------
// Kernel 1: kernel_max[g,p,n] = max_m  sum_c x[g,c,m] * x[p,c,n]
// Grid: (64 row-blocks of 96, 32 g).  Block: 192 threads = 6 waves.
// LDS tiles are K-minor: As[r][kc], Bs[m][kc] with stride 36 so every
// operand fragment is one aligned ds_load_b64.
// ---------------------------------------------------------------------------
__global__ __launch_bounds__(192) void gram_max_kernel(const float* __restrict__ x,
                                                       float* __restrict__ out) {
    __shared__ float As[96 * LSTRIDE];    // As[r*36 + kc]  = x[p, c0+kc, nBase + r]
    __shared__ float Bs[HW * LSTRIDE];    // Bs[m*36 + kc]  = x[g, c0+kc, m]

    const int tid  = threadIdx.x;
    const int lane = tid & 31;
    const int wave = tid >> 5;        // 0..5
    const int half = lane >> 4;       // 0/1 -> K offset 0/2
    const int pos  = lane & 15;       // M (A) / N (B)

    const int g         = blockIdx.y;
    const int wgRowBase = blockIdx.x * 96;     // 0..6048, step 96
    const int p         = wgRowBase / HW;      // source batch for A rows
    const int wgNBase   = wgRowBase % HW;      // 0 or 96

    const float* __restrict__ srcB = x + (size_t)g * CC * HW;
    const float* __restrict__ srcA = x + (size_t)p * CC * HW + wgNBase;

    v8f acc[12];
    #pragma unroll
    for (int t = 0; t < 12; ++t) {
        #pragma unroll
        for (int v = 0; v < 8; ++v) acc[t][v] = 0.0f;
    }

    for (int c0 = 0; c0 < CC; c0 += KCH) {
        __syncthreads();
        // ---- Stage B chunk: 32 x 192 floats = 1536 float4s, 8 per thread ----
        {
            v4f tb[8];
            #pragma unroll
            for (int j = 0; j < 8; ++j) {
                int f  = j * 192 + tid;          // float4 index
                int kc = f / 48;                 // 48 float4 per k-row
                int mm = (f - kc * 48) * 4;
                tb[j] = *(const v4f*)&srcB[(size_t)(c0 + kc) * HW + mm];
            }
            #pragma unroll
            for (int j = 0; j < 8; ++j) {
                int f  = j * 192 + tid;
                int kc = f / 48;
                int mm = (f - kc * 48) * 4;
                #pragma unroll
                for (int u = 0; u < 4; ++u)
                    Bs[(mm + u) * LSTRIDE + kc] = tb[j][u];
            }
        }
        // ---- Stage A chunk: 32 x 96 floats = 768 float4s, 4 per thread ----
        {
            v4f ta[4];
            #pragma unroll
            for (int j = 0; j < 4; ++j) {
                int f  = j * 192 + tid;
                int kc = f / 24;                 // 24 float4 per k-row
                int rr = (f - kc * 24) * 4;
                ta[j] = *(const v4f*)&srcA[(size_t)(c0 + kc) * HW + rr];
            }
            #pragma unroll
            for (int j = 0; j < 4; ++j) {
                int f  = j * 192 + tid;
                int kc = f / 24;
                int rr = (f - kc * 24) * 4;
                #pragma unroll
                for (int u = 0; u < 4; ++u)
                    As[(rr + u) * LSTRIDE + kc] = ta[j][u];
            }
        }
        __syncthreads();

        const float* aRow = &As[(wave * 16 + pos) * LSTRIDE + 2 * half];
        const float* bRow = &Bs[pos * LSTRIDE + 2 * half];
        #pragma unroll
        for (int kq = 0; kq < KCH; kq += 4) {
            v2f a = *(const v2f*)(aRow + kq);               // ds_load_b64
            #pragma unroll
            for (int t = 0; t < 12; ++t) {
                v2f b = *(const v2f*)(bRow + t * 16 * LSTRIDE + kq);  // ds_load_b64
                acc[t] = wmma_f32_k4(a, b, acc[t]);
            }
        }
    }

    // Elementwise max across the 12 N-tiles.
    v8f red = acc[0];
    #pragma unroll
    for (int t = 1; t < 12; ++t)
        #pragma unroll
        for (int v = 0; v < 8; ++v) red[v] = fmaxf(red[v], acc[t][v]);

    // Reduce over N=lane within each 16-lane half (masks < 16 keep halves apart).
    #pragma unroll
    for (int m = 8; m >= 1; m >>= 1)
        #pragma unroll
        for (int v = 0; v < 8; ++v)
            red[v] = fmaxf(red[v], __shfl_xor(red[v], m, 32));

    // D layout: VGPR v holds row M=v (lanes 0-15) and M=v+8 (lanes 16-31).
    float* o = out + (size_t)g * (BB * HW) + wgRowBase + wave * 16 + half * 8;
    if (pos == 0) {
        v4f s0 = {red[0], red[1], red[2], red[3]};
        v4f s1 = {red[4], red[5], red[6], red[7]};
        *(v4f*)(o)     = s0;
        *(v4f*)(o + 4) = s1;
    }
}

// ---------------------------------------------------------------------------
// Kernel 2: projections q = Wq*xf + bq, k = Wk*xf + bk  into d_ws.
// ws layout (m-major, qc-minor): [which][b][m][qc] -> ws[((which*32+b)*192 + m)*64 + qc]
// Grid: (12 n-tiles, 32 b, 2 which).  Block: 128 threads = 4 waves (M-tiles).
// ---------------------------------------------------------------------------
__global__ __launch_bounds__(128) void pam_proj_kernel(const float* __restrict__ x,
                                                       const float* __restrict__ Wq,
                                                       const float* __restrict__ bq,
                                                       const float* __restrict__ Wk,
                                                       const float* __restrict__ bk,
                                                       float* __restrict__ ws) {
    const int tid  = threadIdx.x;
    const int lane = tid & 31;
    const int wave = tid >> 5;        // M-tile 0..3
    const int half = lane >> 4;
    const int pos  = lane & 15;

    const int ntile = blockIdx.x;     // 0..11
    const int b     = blockIdx.y;     // 0..31
    const int which = blockIdx.z;     // 0=q, 1=k

    const float* __restrict__ W    = which ? Wk : Wq;
    const float* __restrict__ bias = which ? bk : bq;
    float* __restrict__ dst = ws + ((size_t)which * BB + b) * (HW * C8);

    const float* __restrict__ wrow = W + (size_t)(wave * 16 + pos) * CC + 2 * half;   // A: M=pos
    const float* __restrict__ xb   = x + (size_t)b * CC * HW + ntile * 16 + pos;      // B: N=pos

    v8f acc;
    #pragma unroll
    for (int v = 0; v < 8; ++v) acc[v] = 0.0f;

    #pragma unroll 4
    for (int c0 = 0; c0 < CC; c0 += 4) {
        v2f a = *(const v2f*)(wrow + c0);       // global b64 (consecutive c)
        v2f bf;
        bf.x = xb[(size_t)(c0 + 2 * half + 0) * HW];
        bf.y = xb[(size_t)(c0 + 2 * half + 1) * HW];
        acc = wmma_f32_k4(a, bf, acc);
    }

    // D: row m = ntile*16 + pos (N), channel qc = wave*16 + v + 8*half (M).
    {
        const int qc0 = wave * 16 + 8 * half;
        float* drow = dst + (size_t)(ntile * 16 + pos) * C8 + qc0;
        v4f s0, s1;
        #pragma unroll
        for (int v = 0; v < 4; ++v) {
            s0[v] = acc[v]     + bias[qc0 + v];
            s1[v] = acc[v + 4] + bias[qc0 + v + 4];
        }
        *(v4f*)(drow)     = s0;                 // global b128 x2
        *(v4f*)(drow + 4) = s1;
    }
}

// ---------------------------------------------------------------------------
// Kernel 3: pos_max[b,m] = max_n  sum_qc q[b,qc,m]*k[b,qc,n]   (K = 64)
// Grid: (12 m-tiles, 32 b).  Block: one wave.  A fragments hoisted (reused 12x).
// ---------------------------------------------------------------------------
__global__ __launch_bounds__(32) void pam_energy_kernel(const float* __restrict__ ws,
                                                        float* __restrict__ out) {
    const int lane = threadIdx.x & 31;
    const int half = lane >> 4;
    const int pos  = lane & 15;
    const int mtile = blockIdx.x;     // 0..11
    const int b     = blockIdx.y;     // 0..31

    const float* __restrict__ q = ws + (size_t)b * (HW * C8);
    const float* __restrict__ k = ws + (size_t)(BB + b) * (HW * C8);

    // Hoist all 16 A fragments for this m-tile: a[i] covers K = 4i + {0,1} (+2 upper half).
    v2f afr[16];
    {
        const float* qrow = q + (size_t)(mtile * 16 + pos) * C8 + 2 * half;
        #pragma unroll
        for (int i = 0; i < 16; ++i)
            afr[i] = *(const v2f*)(qrow + 4 * i);          // global b64
    }

    v8f red;
    #pragma unroll
    for (int v = 0; v < 8; ++v) red[v] = -INFINITY;

    #pragma unroll 2
    for (int nt = 0; nt < 12; ++nt) {
        const float* krow = k + (size_t)(nt * 16 + pos) * C8 + 2 * half;
        v8f acc;
        #pragma unroll
        for (int v = 0; v < 8; ++v) acc[v] = 0.0f;
        #pragma unroll
        for (int i = 0; i < 16; ++i) {
            v2f bf = *(const v2f*)(krow + 4 * i);          // global b64
            acc = wmma_f32_k4(afr[i], bf, acc);
        }
        #pragma unroll
        for (int v = 0; v < 8; ++v) red[v] = fmaxf(red[v], acc[v]);
    }

    #pragma unroll
    for (int m = 8; m >= 1; m >>= 1)
        #pragma unroll
        for (int v = 0; v < 8; ++v)
            red[v] = fmaxf(red[v], __shfl_xor(red[v], m, 32));

    float* o = out + (size_t)BB * (BB * HW) + (size_t)b * HW + mtile * 16 + half * 8;
    if (pos == 0) {
        v4f s0 = {red[0], red[1], red[2], red[3]};
        v4f s1 = {red[4], red[5], red[6], red[7]};
        *(v4f*)(o)     = s0;
        *(v4f*)(o + 4) = s1;
    }
}

// ---------------------------------------------------------------------------
extern "C" void kernel_launch(void* const* d_in, const int* in_sizes, int n_in,
                              void* d_out, int out_size, void* d_ws, size_t ws_size,
                              hipStream_t stream) {
    const float* x  = (const float*)d_in[0];   // [32,512,24,8]
    const float* Wq = (const float*)d_in[1];   // [64,512]
    const float* bq = (const float*)d_in[2];   // [64]
    const float* Wk = (const float*)d_in[3];   // [64,512]
    const float* bk = (const float*)d_in[4];   // [64]
    float* out = (float*)d_out;                // [33,32,192]
    float* ws  = (float*)d_ws;                 // needs 2*32*192*64*4 = 3 MB

    // Gram + row-max: 32 groups x 64 row-blocks of 96 rows.
    gram_max_kernel<<<dim3(64, 32), 192, 0, stream>>>(x, out);
    // PAM projections into workspace, then energy + max.
    pam_proj_kernel<<<dim3(12, 32, 2), 128, 0, stream>>>(x, Wq, bq, Wk, bk, ws);
    pam_energy_kernel<<<dim3(12, 32), 32, 0, stream>>>(ws, out);
}